// PC_Dipole_damped_electrostatics_48241072668729
// MI455X (gfx1250) — compile-verified
//
#include <hip/hip_runtime.h>

// ---------------------------------------------------------------------------
// PC dipole damped electrostatics: per-edge energy + segment_sum over idx_i.
//
// Roofline (MI455X): 76.8 MB one-shot edge stream -> ~3.3 us floor @ 23.3 TB/s.
// Gather tables (charges+dipoles, 800 KB) are L2-resident (192 MB L2), so the
// random gathers are request-rate bound -> pack (q,d) into float2 to halve
// request count. Scatter = 6.4M native global_atomic_add_f32. No matrix
// structure -> WMMA inapplicable; gfx1250 paths used: wave32 sizing, TH=NT
// b128 loads, global_prefetch_b8, HW f32 atomics.
// ---------------------------------------------------------------------------

typedef float v4f __attribute__((ext_vector_type(4)));
typedef int   v4i __attribute__((ext_vector_type(4)));
typedef float v2f __attribute__((ext_vector_type(2)));

#define KEHALF          7.199822675975274f
#define CUTOFF          12.0f
#define CUTOFF_SR       2.0f
#define INV_CUTOFF_SQ   (1.0f / 144.0f)   // 1/CUTOFF^2
#define TWO_OVER_CUTOFF (1.0f / 6.0f)     // 2/CUTOFF

__device__ __forceinline__ float edge_energy(float d, float qq, float dd) {
    // shielded distance
    float d_sh    = sqrtf(fmaf(d, d, 1.0f));
    float inv_d   = 1.0f / d;
    float inv_dsh = 1.0f / d_sh;

    // poly6 switch on [0, CUTOFF_SR): 1 - 10x^3 + 15x^4 - 6x^5, x = d/2
    float x  = d * (1.0f / CUTOFF_SR);
    float x3 = x * x * x;
    float sw = fmaf(x3, fmaf(x, fmaf(-6.0f, x, 15.0f), -10.0f), 1.0f);
    float sw_off = (d < CUTOFF_SR) ? sw : 0.0f;
    float sw_on  = 1.0f - sw_off;

    // shifted-force Coulomb (charge-charge)
    float E_sh_q  = inv_dsh + fmaf(d_sh, INV_CUTOFF_SQ, -TWO_OVER_CUTOFF);
    float E_ord_q = inv_d   + fmaf(d,    INV_CUTOFF_SQ, -TWO_OVER_CUTOFF);
    float Eq = qq * fmaf(sw_off, E_sh_q, sw_on * E_ord_q);

    // dipole-dipole (~1/d^3)
    float E_sh_d  = inv_dsh * inv_dsh * inv_dsh;
    float E_ord_d = inv_d   * inv_d   * inv_d;
    float Ed = dd * fmaf(sw_off, E_sh_d, sw_on * E_ord_d);

    float E = KEHALF * (Eq + Ed);
    return (d <= CUTOFF) ? E : 0.0f;
}

__global__ void zero_out_kernel(float* __restrict__ out, int n) {
    int i = blockIdx.x * blockDim.x + threadIdx.x;
    if (i < n) out[i] = 0.0f;
}

// Interleave (charge, dipole) -> float2 table so each per-edge gather is a
// single b64 load instead of two b32 loads (halves random-request count).
__global__ void pack_tables_kernel(const float* __restrict__ charges,
                                   const float* __restrict__ dipoles,
                                   v2f* __restrict__ tab, int n) {
    int i = blockIdx.x * blockDim.x + threadIdx.x;
    if (i < n) {
        v2f v;
        v.x = charges[i];
        v.y = dipoles[i];
        tab[i] = v;
    }
}

__global__ __launch_bounds__(256)
void pc_dipole_edge_packed_kernel(const v2f* __restrict__ tab,
                                  const float* __restrict__ dist,
                                  const int*   __restrict__ idx_i,
                                  const int*   __restrict__ idx_j,
                                  float*       __restrict__ out,
                                  int n_edges) {
    const int nvec   = n_edges >> 2;            // 4 edges per thread-iteration
    const int stride = gridDim.x * blockDim.x;
    const int tid0   = blockIdx.x * blockDim.x + threadIdx.x;

    const v4f* dvec = (const v4f*)dist;
    const v4i* ivec = (const v4i*)idx_i;
    const v4i* jvec = (const v4i*)idx_j;

    for (int t = tid0; t < nvec; t += stride) {
        // prefetch next grid-stride chunk (global_prefetch_b8)
        int tn = t + stride;
        if (tn < nvec) {
            __builtin_prefetch((const void*)(dvec + tn), 0, 0);
            __builtin_prefetch((const void*)(ivec + tn), 0, 0);
            __builtin_prefetch((const void*)(jvec + tn), 0, 0);
        }

        // one-shot 77MB stream: TH=NT b128 loads keep the 800KB gather table
        // resident in near caches
        v4f d4 = __builtin_nontemporal_load(dvec + t);
        v4i i4 = __builtin_nontemporal_load(ivec + t);
        v4i j4 = __builtin_nontemporal_load(jvec + t);

#pragma unroll
        for (int k = 0; k < 4; ++k) {
            float d  = d4[k];
            int   ai = i4[k];
            int   aj = j4[k];
            v2f   ti = tab[ai];                 // single b64 gather per atom
            v2f   tj = tab[aj];
            float qq = ti.x * tj.x;
            float dd = ti.y * tj.y;
            float E  = edge_energy(d, qq, dd);
            unsafeAtomicAdd(&out[ai], E);       // global_atomic_add_f32
        }
    }

    // tail (n_edges not a multiple of 4)
    int rem = n_edges & 3;
    if (rem != 0 && tid0 < rem) {
        int e    = (nvec << 2) + tid0;
        float d  = dist[e];
        v2f   ti = tab[idx_i[e]];
        v2f   tj = tab[idx_j[e]];
        unsafeAtomicAdd(&out[idx_i[e]], edge_energy(d, ti.x * tj.x, ti.y * tj.y));
    }
}

// Fallback (workspace too small for the packed table): 4x b32 gathers.
__global__ __launch_bounds__(256)
void pc_dipole_edge_kernel(const float* __restrict__ charges,
                           const float* __restrict__ dipoles,
                           const float* __restrict__ dist,
                           const int*   __restrict__ idx_i,
                           const int*   __restrict__ idx_j,
                           float*       __restrict__ out,
                           int n_edges) {
    const int nvec   = n_edges >> 2;
    const int stride = gridDim.x * blockDim.x;
    const int tid0   = blockIdx.x * blockDim.x + threadIdx.x;

    const v4f* dvec = (const v4f*)dist;
    const v4i* ivec = (const v4i*)idx_i;
    const v4i* jvec = (const v4i*)idx_j;

    for (int t = tid0; t < nvec; t += stride) {
        int tn = t + stride;
        if (tn < nvec) {
            __builtin_prefetch((const void*)(dvec + tn), 0, 0);
            __builtin_prefetch((const void*)(ivec + tn), 0, 0);
            __builtin_prefetch((const void*)(jvec + tn), 0, 0);
        }
        v4f d4 = __builtin_nontemporal_load(dvec + t);
        v4i i4 = __builtin_nontemporal_load(ivec + t);
        v4i j4 = __builtin_nontemporal_load(jvec + t);
#pragma unroll
        for (int k = 0; k < 4; ++k) {
            float d  = d4[k];
            int   ai = i4[k];
            int   aj = j4[k];
            float qq = charges[ai] * charges[aj];
            float dd = dipoles[ai] * dipoles[aj];
            unsafeAtomicAdd(&out[ai], edge_energy(d, qq, dd));
        }
    }
    int rem = n_edges & 3;
    if (rem != 0 && tid0 < rem) {
        int e = (nvec << 2) + tid0;
        float qq = charges[idx_i[e]] * charges[idx_j[e]];
        float dd = dipoles[idx_i[e]] * dipoles[idx_j[e]];
        unsafeAtomicAdd(&out[idx_i[e]], edge_energy(dist[e], qq, dd));
    }
}

extern "C" void kernel_launch(void* const* d_in, const int* in_sizes, int n_in,
                              void* d_out, int out_size, void* d_ws, size_t ws_size,
                              hipStream_t stream) {
    const float* charges = (const float*)d_in[0];
    const float* dipoles = (const float*)d_in[1];
    const float* dist    = (const float*)d_in[2];
    const int*   idx_i   = (const int*)d_in[3];
    const int*   idx_j   = (const int*)d_in[4];
    float*       out     = (float*)d_out;

    const int n_atoms = in_sizes[0];
    const int n_edges = in_sizes[2];

    // 1) zero the accumulation buffer (deterministic, graph-capture safe)
    int zb = (n_atoms + 255) / 256;
    zero_out_kernel<<<zb, 256, 0, stream>>>(out, n_atoms);

    const int blocks = 2048;   // x 256 threads = 8 wave32 per block

    if (ws_size >= (size_t)n_atoms * sizeof(v2f)) {
        // 2) pack (charge, dipole) -> float2 table in workspace (~1.6MB traffic)
        v2f* tab = (v2f*)d_ws;
        pack_tables_kernel<<<zb, 256, 0, stream>>>(charges, dipoles, tab, n_atoms);
        // 3) edge stream with b64 gathers
        pc_dipole_edge_packed_kernel<<<blocks, 256, 0, stream>>>(
            tab, dist, idx_i, idx_j, out, n_edges);
    } else {
        pc_dipole_edge_kernel<<<blocks, 256, 0, stream>>>(
            charges, dipoles, dist, idx_i, idx_j, out, n_edges);
    }
}